// MultiBoxLoss_2963527434602
// MI455X (gfx1250) — compile-verified
//
#include <hip/hip_runtime.h>
#include <math.h>

#define BB 64
#define PP 16800
#define OO 64
#define THRESH   0.35f
#define VALID_TH 0.2f
#define VAR0 0.1f
#define VAR1 0.2f

typedef __attribute__((ext_vector_type(2))) float v2f;
typedef __attribute__((ext_vector_type(8))) float v8f;

// Exact f32 wave32 sum using V_WMMA_F32_16X16X4_F32.
// Lane L contributes A[L%16, 2*(L/16)] = v (other K entries 0), B = ones(4x16)
// => D[m,n] = v_m + v_{m+16}.  Sum of the 8 D VGPRs gives half-sums in each
// 16-lane group; one shfl_xor(16) completes the 32-lane total in all lanes.
__device__ __forceinline__ float wave_sum_wmma(float v) {
    v2f a; a[0] = v;    a[1] = 0.0f;
    v2f b; b[0] = 1.0f; b[1] = 1.0f;
    v8f c = {};
    c = __builtin_amdgcn_wmma_f32_16x16x4_f32(false, a, false, b, (short)0, c, false, false);
    float s = c[0] + c[1] + c[2] + c[3] + c[4] + c[5] + c[6] + c[7];
    s += __shfl_xor(s, 16, 32);
    return s;
}

__device__ __forceinline__ float sl1(float x) {
    float a = fabsf(x);
    return a < 1.0f ? 0.5f * a * a : a - 0.5f;
}

// ---------------- K0: zero small accumulators ----------------
__global__ void k_init(unsigned long long* bp, float* batch_npos, float* batch_topk,
                       int* any_valid, float* acc) {
    int i = blockIdx.x * blockDim.x + threadIdx.x;
    if (i < BB * OO) bp[i] = 0ull;
    if (i < BB) { batch_npos[i] = 0.0f; batch_topk[i] = 0.0f; any_valid[i] = 0; }
    if (i < 4)  acc[i] = 0.0f;
}

// ---------------- K1: best truth per prior (max+argmax over O) ----------------
__global__ void k_best_truth(const float* __restrict__ priors,
                             const float* __restrict__ targets,
                             float* __restrict__ bt_ov, int* __restrict__ bt_idx) {
    __shared__ float tx1[OO], ty1[OO], tx2[OO], ty2[OO], tarea[OO];
    const int b = blockIdx.y;
    const int p = blockIdx.x * blockDim.x + threadIdx.x;
    if (threadIdx.x < OO) {
        const float* t = targets + (size_t)(b * OO + threadIdx.x) * 15;
        float x1 = t[0], y1 = t[1], x2 = t[2], y2 = t[3];
        tx1[threadIdx.x] = x1; ty1[threadIdx.x] = y1;
        tx2[threadIdx.x] = x2; ty2[threadIdx.x] = y2;
        tarea[threadIdx.x] = (x2 - x1) * (y2 - y1);
    }
    if (p < PP) __builtin_prefetch(priors + (size_t)p * 4, 0, 0); // global_prefetch_b8
    __syncthreads();
    if (p >= PP) return;
    const float4 pr = *(const float4*)(priors + (size_t)p * 4); // cx,cy,w,h
    const float px1 = pr.x - 0.5f * pr.z, py1 = pr.y - 0.5f * pr.w;
    const float px2 = pr.x + 0.5f * pr.z, py2 = pr.y + 0.5f * pr.w;
    const float parea = pr.z * pr.w;
    float best = -1.0f; int bi = 0;
    #pragma unroll 4
    for (int t = 0; t < OO; ++t) {
        float lx = fmaxf(px1, tx1[t]), ly = fmaxf(py1, ty1[t]);
        float rx = fminf(px2, tx2[t]), ry = fminf(py2, ty2[t]);
        float w = fmaxf(rx - lx, 0.0f), h = fmaxf(ry - ly, 0.0f);
        float inter = w * h;
        float iou = inter / (tarea[t] + parea - inter);
        if (iou > best) { best = iou; bi = t; }   // strict > : first index on tie
    }
    bt_ov[b * PP + p] = best;
    bt_idx[b * PP + p] = bi;
}

// ---------------- K2: best prior per truth (packed u64 max) ----------------
__global__ void k_best_prior(const float* __restrict__ priors,
                             const float* __restrict__ targets,
                             unsigned long long* __restrict__ bp) {
    const int b = blockIdx.y, o = blockIdx.x, tid = threadIdx.x;
    const float* t = targets + (size_t)(b * OO + o) * 15;
    const float x1 = t[0], y1 = t[1], x2 = t[2], y2 = t[3];
    const float ta = (x2 - x1) * (y2 - y1);
    unsigned long long best = 0ull;
    for (int p = tid; p < PP; p += 256) {
        const float4 pr = *(const float4*)(priors + (size_t)p * 4);
        float px1 = pr.x - 0.5f * pr.z, py1 = pr.y - 0.5f * pr.w;
        float px2 = pr.x + 0.5f * pr.z, py2 = pr.y + 0.5f * pr.w;
        float lx = fmaxf(px1, x1), ly = fmaxf(py1, y1);
        float rx = fminf(px2, x2), ry = fminf(py2, y2);
        float w = fmaxf(rx - lx, 0.0f), h = fmaxf(ry - ly, 0.0f);
        float inter = w * h;
        float iou = inter / (ta + pr.z * pr.w - inter);       // >= 0, bits monotone
        unsigned long long pk = ((unsigned long long)__float_as_uint(iou) << 32)
                              | (unsigned long long)(0xFFFFFFFFu - (unsigned)p);
        if (pk > best) best = pk;                              // smallest p on tie
    }
    __shared__ unsigned long long sm[256];
    sm[tid] = best; __syncthreads();
    for (int off = 128; off > 0; off >>= 1) {
        if (tid < off && sm[tid + off] > sm[tid]) sm[tid] = sm[tid + off];
        __syncthreads();
    }
    if (tid == 0) bp[b * OO + o] = sm[0];
}

// ---------------- K3: sequential scatter override (matches JAX last-wins) ----------------
__global__ void k_override(const unsigned long long* __restrict__ bp,
                           float* __restrict__ bt_ov, int* __restrict__ bt_idx,
                           int* __restrict__ any_valid) {
    __shared__ int   s_idx[BB * OO];
    __shared__ float s_val[BB * OO];
    const int b = threadIdx.x;
    if (b >= BB) return;
    int any = 0;
    for (int o = 0; o < OO; ++o) {            // gather BEFORE any writes
        unsigned long long pk = bp[b * OO + o];
        float ov = __uint_as_float((unsigned)(pk >> 32));
        int idx = (int)(0xFFFFFFFFu - (unsigned)(pk & 0xFFFFFFFFull));
        int valid = ov >= VALID_TH;
        s_idx[b * OO + o] = idx;
        s_val[b * OO + o] = valid ? 2.0f : bt_ov[b * PP + idx];
        any |= valid;
    }
    for (int o = 0; o < OO; ++o) {            // sequential writes: last wins
        int idx = s_idx[b * OO + o];
        bt_ov[b * PP + idx]  = s_val[b * OO + o];
        bt_idx[b * PP + idx] = o;
    }
    any_valid[b] = any;
}

// ---------------- K4: encode + losses + rank scores (WMMA wave reductions) ----------------
__global__ void k_losses(const float* __restrict__ loc_data, const float* __restrict__ conf_data,
                         const float* __restrict__ landm_data, const float* __restrict__ priors,
                         const float* __restrict__ targets,
                         const float* __restrict__ bt_ov, const int* __restrict__ bt_idx,
                         const int* __restrict__ any_valid,
                         unsigned* __restrict__ rank_bits,
                         float* __restrict__ batch_npos, float* __restrict__ acc) {
    __shared__ float st[OO * 15];
    const int b = blockIdx.y;
    for (int i = threadIdx.x; i < OO * 15; i += blockDim.x)
        st[i] = targets[(size_t)b * OO * 15 + i];
    __syncthreads();

    const int p  = blockIdx.x * blockDim.x + threadIdx.x;
    const bool inb = p < PP;
    const int pp = inb ? p : PP - 1;

    const float4 pr = *(const float4*)(priors + (size_t)pp * 4);
    const float ovp = bt_ov[b * PP + pp];
    const int   j   = bt_idx[b * PP + pp];
    const int   av  = any_valid[b];

    const float label = st[j * 15 + 14];
    const int conf = (!inb || !av || ovp < THRESH) ? 0 : (int)label;
    const bool pos  = conf != 0;
    const bool pos1 = conf > 0;

    float loss_l = 0.0f, loss_lm = 0.0f;
    if (pos) {
        float mx1 = st[j * 15 + 0], my1 = st[j * 15 + 1];
        float mx2 = st[j * 15 + 2], my2 = st[j * 15 + 3];
        float gcx = ((mx1 + mx2) * 0.5f - pr.x) / (VAR0 * pr.z);
        float gcy = ((my1 + my2) * 0.5f - pr.y) / (VAR0 * pr.w);
        float gw  = __logf((mx2 - mx1) / pr.z) / VAR1;
        float gh  = __logf((my2 - my1) / pr.w) / VAR1;
        const float4 ld = *(const float4*)(loc_data + (size_t)(b * PP + pp) * 4);
        loss_l = sl1(ld.x - gcx) + sl1(ld.y - gcy) + sl1(ld.z - gw) + sl1(ld.w - gh);
    }
    if (pos1) {
        const float* lm = landm_data + (size_t)(b * PP + pp) * 10;
        #pragma unroll
        for (int q = 0; q < 5; ++q) {
            float ltx = (st[j * 15 + 4 + 2 * q] - pr.x) / (VAR0 * pr.z);
            float lty = (st[j * 15 + 5 + 2 * q] - pr.y) / (VAR0 * pr.w);
            loss_lm += sl1(lm[2 * q] - ltx) + sl1(lm[2 * q + 1] - lty);
        }
    }
    const float c0 = conf_data[(size_t)(b * PP + pp) * 2 + 0];
    const float c1 = conf_data[(size_t)(b * PP + pp) * 2 + 1];
    const float m  = fmaxf(c0, c1);
    const float lse = m + __logf(__expf(c0 - m) + __expf(c1 - m));
    const float ce  = lse - (pos ? c1 : c0);
    if (inb) rank_bits[b * PP + p] = __float_as_uint(pos ? 0.0f : ce);

    float pos_ce = pos  ? ce   : 0.0f;
    float npos   = pos  ? 1.0f : 0.0f;
    float n1     = pos1 ? 1.0f : 0.0f;

    // control flow is reconverged here: EXEC all 1s for WMMA
    float s;
    s = wave_sum_wmma(loss_l);  if ((threadIdx.x & 31) == 0) atomicAdd(&acc[0], s);
    s = wave_sum_wmma(loss_lm); if ((threadIdx.x & 31) == 0) atomicAdd(&acc[1], s);
    s = wave_sum_wmma(pos_ce);  if ((threadIdx.x & 31) == 0) atomicAdd(&acc[2], s);
    s = wave_sum_wmma(n1);      if ((threadIdx.x & 31) == 0) atomicAdd(&acc[3], s);
    s = wave_sum_wmma(npos);    if ((threadIdx.x & 31) == 0) atomicAdd(&batch_npos[b], s);
}

// ---------------- K5: per-row exact top-k sum via radix select ----------------
__global__ void k_negsum(const unsigned* __restrict__ rank_bits,
                         const float* __restrict__ batch_npos,
                         float* __restrict__ batch_topk) {
    const int b = blockIdx.x, tid = threadIdx.x;
    __shared__ unsigned hist[256];
    __shared__ unsigned s_prefix, s_krem;
    __shared__ float    ssum[256];

    const int npos = (int)(batch_npos[b] + 0.5f);
    long long k = 7LL * (long long)npos;
    if (k > PP - 1) k = PP - 1;
    if (k <= 0) { if (tid == 0) batch_topk[b] = 0.0f; return; }

    const unsigned* row = rank_bits + (size_t)b * PP;
    if (tid == 0) { s_prefix = 0u; s_krem = (unsigned)k; }
    __syncthreads();

    for (int d = 3; d >= 0; --d) {
        hist[tid] = 0u;
        __syncthreads();
        const unsigned prefix = s_prefix;
        const unsigned hm = (d == 3) ? 0u : (0xFFFFFFFFu << (8 * (d + 1)));
        const int shift = 8 * d;
        for (int i = tid; i < PP; i += 256) {
            unsigned v = row[i];
            if ((v & hm) == prefix) atomicAdd(&hist[(v >> shift) & 0xFFu], 1u);
        }
        __syncthreads();
        if (tid == 0) {
            unsigned krem = s_krem, cum = 0u; int sel = 0;
            for (int bin = 255; bin >= 0; --bin) {
                unsigned c = hist[bin];
                if (cum + c >= krem) { sel = bin; break; }
                cum += c;
            }
            s_prefix = prefix | ((unsigned)sel << shift);
            s_krem = krem - cum;      // after last pass: k - count(v > threshold)
        }
        __syncthreads();
    }
    const unsigned tbits = s_prefix;
    float lsum = 0.0f;
    for (int i = tid; i < PP; i += 256) {
        unsigned v = row[i];
        if (v > tbits) lsum += __uint_as_float(v);
    }
    ssum[tid] = lsum; __syncthreads();
    for (int off = 128; off > 0; off >>= 1) {
        if (tid < off) ssum[tid] += ssum[tid + off];
        __syncthreads();
    }
    if (tid == 0)
        batch_topk[b] = ssum[0] + (float)s_krem * __uint_as_float(tbits);
}

// ---------------- K6: finalize ----------------
__global__ void k_final(const float* __restrict__ acc, const float* __restrict__ batch_npos,
                        const float* __restrict__ batch_topk, float* __restrict__ out) {
    float npos_total = 0.0f, negsum = 0.0f;
    for (int b = 0; b < BB; ++b) { npos_total += batch_npos[b]; negsum += batch_topk[b]; }
    float N  = fmaxf(npos_total, 1.0f);
    float N1 = fmaxf(acc[3], 1.0f);
    out[0] = acc[0] / N;                 // loss_l / N
    out[1] = (acc[2] + negsum) / N;      // loss_c / N
    out[2] = acc[1] / N1;                // loss_landm / N1
}

extern "C" void kernel_launch(void* const* d_in, const int* in_sizes, int n_in,
                              void* d_out, int out_size, void* d_ws, size_t ws_size,
                              hipStream_t stream) {
    const float* loc_data   = (const float*)d_in[0];   // B*P*4
    const float* conf_data  = (const float*)d_in[1];   // B*P*2
    const float* landm_data = (const float*)d_in[2];   // B*P*10
    const float* priors     = (const float*)d_in[3];   // P*4
    const float* targets    = (const float*)d_in[4];   // B*O*15

    char* ws = (char*)d_ws;
    float*    bt_ov     = (float*)ws;                                    // B*P
    int*      bt_idx    = (int*)(ws + 4ull * BB * PP);                   // B*P
    unsigned* rank_bits = (unsigned*)(ws + 8ull * BB * PP);              // B*P
    unsigned long long* bp = (unsigned long long*)(ws + 12ull * BB * PP);// B*O (8B aligned)
    float* batch_npos = (float*)(ws + 12ull * BB * PP + 8ull * BB * OO);
    float* batch_topk = batch_npos + BB;
    int*   any_valid  = (int*)(batch_topk + BB);
    float* acc        = (float*)(any_valid + BB);                        // 4 floats

    k_init<<<(BB * OO + 255) / 256, 256, 0, stream>>>(bp, batch_npos, batch_topk, any_valid, acc);

    dim3 g1((PP + 255) / 256, BB);
    k_best_truth<<<g1, 256, 0, stream>>>(priors, targets, bt_ov, bt_idx);

    dim3 g2(OO, BB);
    k_best_prior<<<g2, 256, 0, stream>>>(priors, targets, bp);

    k_override<<<1, BB, 0, stream>>>(bp, bt_ov, bt_idx, any_valid);

    k_losses<<<g1, 256, 0, stream>>>(loc_data, conf_data, landm_data, priors, targets,
                                     bt_ov, bt_idx, any_valid, rank_bits, batch_npos, acc);

    k_negsum<<<BB, 256, 0, stream>>>(rank_bits, batch_npos, batch_topk);

    k_final<<<1, 1, 0, stream>>>(acc, batch_npos, batch_topk, (float*)d_out);
}